// CrossNetMix_48163763257624
// MI455X (gfx1250) — compile-verified
//
#include <hip/hip_runtime.h>
#include <hip/hip_bf16.h>

typedef float v2f __attribute__((ext_vector_type(2)));
typedef float v8f __attribute__((ext_vector_type(8)));

#define B_TOT    16384
#define D_DIM    1024
#define E_EXP    4
#define R_RANK   32
#define N1       128          // E*R
#define L_LAYERS 2
#define B_ROWS   32           // batch rows per workgroup
#define THREADS  256          // 8 wave32
#define XPITCH   1028         // 1024 + 4 pad: lane-strided column reads hit distinct banks
#define YPITCH   132          // 128 + 4 pad

// LDS layout (floats): X0[32][XPITCH], XL[32][XPITCH], YB[32][YPITCH], CB[32][YPITCH],
//                      GT[128] (gates), GP[256] (gate partials)
#define SMEM_FLOATS (2 * B_ROWS * XPITCH + 2 * B_ROWS * YPITCH + B_ROWS * E_EXP + THREADS)

// Branch-free tanh: 1 - 2/(exp(2x)+1).  exp(2x)->inf => 1 ; ->0 => -1.
// Uses v_exp_f32 + fast reciprocal; no EXEC-divergent range reduction like libm tanhf.
__device__ __forceinline__ float fast_tanh(float x) {
    const float e = __expf(2.0f * x);
    return 1.0f - 2.0f * __frcp_rn(e + 1.0f);
}

extern "C" __global__ __launch_bounds__(THREADS)
void crossnet_mix_kernel(const float* __restrict__ x,
                         const float* __restrict__ U,
                         const float* __restrict__ V,
                         const float* __restrict__ C,
                         const float* __restrict__ bias,
                         const float* __restrict__ gate_w,
                         float* __restrict__ out)
{
    extern __shared__ float smem[];
    float* X0 = smem;
    float* XL = smem + B_ROWS * XPITCH;
    float* YB = smem + 2 * B_ROWS * XPITCH;
    float* CB = YB + B_ROWS * YPITCH;
    float* GT = CB + B_ROWS * YPITCH;
    float* GP = GT + B_ROWS * E_EXP;

    const int tid  = threadIdx.x;
    const int wave = tid >> 5;
    const int lane = tid & 31;
    const int lo16 = lane & 15;
    const int hiK  = lane >> 4;                 // selects K pair {0,1} vs {2,3} (ISA A/B frag layout)
    const int rowBase = blockIdx.x * B_ROWS;

    // ---- Stage x tile into X0 (residual) and XL (running activation) ----
    for (int i = tid; i < B_ROWS * (D_DIM / 4); i += THREADS) {
        const int r  = i >> 8;                  // / (D_DIM/4)
        const int c4 = i & 255;
        const float4 val = ((const float4*)(x + (size_t)(rowBase + r) * D_DIM))[c4];
        *(float4*)&X0[r * XPITCH + c4 * 4] = val;
        *(float4*)&XL[r * XPITCH + c4 * 4] = val;
    }
    __syncthreads();

    const v8f vzero = {};

    for (int l = 0; l < L_LAYERS; ++l) {
        const float* Vl = V + (size_t)l * E_EXP * D_DIM * R_RANK;
        const float* Ul = U + (size_t)l * E_EXP * D_DIM * R_RANK;
        const float* Cl = C + (size_t)l * E_EXP * R_RANK * R_RANK;
        const float* bl = bias + (size_t)l * D_DIM;

        // ================= GEMM1: Y(32x128) = XL(32x1024) @ Vcat(1024x128) ============
        // wave owns a 16-wide column strip; strip sits inside one expert (R=32)
        const int n0 = wave * 16;
        const int e1 = n0 >> 5;
        const int c1 = (n0 & 31) + lo16;        // r-index within expert for this lane
        v8f acc0 = vzero, acc1 = vzero;
        for (int k0 = 0; k0 < D_DIM; k0 += 4) {
            const int kk = k0 + 2 * hiK;
            const v2f a0 = *(const v2f*)&XL[lo16 * XPITCH + kk];          // M-tile 0 rows 0..15
            const v2f a1 = *(const v2f*)&XL[(lo16 + 16) * XPITCH + kk];   // M-tile 1 rows 16..31
            const float* vb = Vl + ((size_t)e1 * D_DIM + kk) * R_RANK + c1;
            v2f b;
            b.x = vb[0];                         // Vcat[kk  ][n0+lo16]
            b.y = vb[R_RANK];                    // Vcat[kk+1][n0+lo16]
            acc0 = __builtin_amdgcn_wmma_f32_16x16x4_f32(false, a0, false, b, (short)0, acc0, false, false);
            acc1 = __builtin_amdgcn_wmma_f32_16x16x4_f32(false, a1, false, b, (short)0, acc1, false, false);
        }
        // v = tanh(Y) -> YB ; C/D layout: VGPR j holds M=j (lanes0-15) / M=j+8 (lanes16-31), N=lane&15
#pragma unroll
        for (int j = 0; j < 8; ++j) {
            const int r0 = j + 8 * hiK;
            YB[r0 * YPITCH + n0 + lo16]        = fast_tanh(acc0[j]);
            YB[(r0 + 16) * YPITCH + n0 + lo16] = fast_tanh(acc1[j]);
        }

        // ================= gate logits: (32 rows) x (4 experts), split over 2 threads ==
        {
            const int b = tid >> 3;              // 32 rows
            const int e = (tid >> 1) & 3;        // 4 experts
            const int h = tid & 1;               // half of D
            const float4* gw = (const float4*)(gate_w + (size_t)e * D_DIM + h * (D_DIM / 2));
            const float* xr = &XL[b * XPITCH + h * (D_DIM / 2)];
            float s = 0.f;
            for (int d4 = 0; d4 < D_DIM / 8; ++d4) {
                const float4 g = gw[d4];
                const float* xp = xr + d4 * 4;
                s += g.x * xp[0] + g.y * xp[1] + g.z * xp[2] + g.w * xp[3];
            }
            GP[tid] = s;
        }
        __syncthreads();                         // covers YB stores + GP partials
        if (tid < B_ROWS) {
            const float g0 = GP[tid * 8 + 0] + GP[tid * 8 + 1];
            const float g1 = GP[tid * 8 + 2] + GP[tid * 8 + 3];
            const float g2 = GP[tid * 8 + 4] + GP[tid * 8 + 5];
            const float g3 = GP[tid * 8 + 6] + GP[tid * 8 + 7];
            const float m  = fmaxf(fmaxf(g0, g1), fmaxf(g2, g3));
            const float e0 = __expf(g0 - m), e1x = __expf(g1 - m);
            const float e2 = __expf(g2 - m), e3 = __expf(g3 - m);
            const float inv = __frcp_rn(e0 + e1x + e2 + e3);
            GT[tid * 4 + 0] = e0 * inv;  GT[tid * 4 + 1] = e1x * inv;
            GT[tid * 4 + 2] = e2 * inv;  GT[tid * 4 + 3] = e3 * inv;
        }
        __syncthreads();

        // ================= c' = gate * tanh(C @ v)  (per-expert R x R) ================
        for (int i = tid; i < B_ROWS * N1; i += THREADS) {
            const int b = i >> 7, col = i & 127;
            const int e = col >> 5, r = col & 31;
            const float* crow = Cl + ((size_t)e * R_RANK + r) * R_RANK;
            const float* vrow = &YB[b * YPITCH + e * R_RANK];
            float s = 0.f;
#pragma unroll 8
            for (int ss = 0; ss < R_RANK; ++ss) s += crow[ss] * vrow[ss];
            CB[b * YPITCH + col] = GT[b * 4 + e] * fast_tanh(s);
        }
        __syncthreads();

        // ================= GEMM2: Z(32x1024) = CB(32x128) @ Ucat(128x1024) ============
        // wave owns 128 output columns => 2 (M) x 8 (N) accumulator tiles
        const int d0 = wave * 128;
        v8f acc2[2][8];
#pragma unroll
        for (int m = 0; m < 2; ++m)
#pragma unroll
            for (int nt = 0; nt < 8; ++nt) acc2[m][nt] = vzero;

        for (int k0 = 0; k0 < N1; k0 += 4) {
            const int kk = k0 + 2 * hiK;
            const int e = kk >> 5, r = kk & 31;  // r even, r+1 stays in same expert block
            const v2f a0 = *(const v2f*)&CB[lo16 * YPITCH + kk];
            const v2f a1 = *(const v2f*)&CB[(lo16 + 16) * YPITCH + kk];
#pragma unroll
            for (int nt = 0; nt < 8; ++nt) {
                const int col = d0 + nt * 16 + lo16;
                const v2f b = *(const v2f*)(Ul + ((size_t)e * D_DIM + col) * R_RANK + r); // contiguous (r fastest)
                acc2[0][nt] = __builtin_amdgcn_wmma_f32_16x16x4_f32(false, a0, false, b, (short)0, acc2[0][nt], false, false);
                acc2[1][nt] = __builtin_amdgcn_wmma_f32_16x16x4_f32(false, a1, false, b, (short)0, acc2[1][nt], false, false);
            }
        }

        // ============ epilogue: XL += X0 * (bias + Z)  (softmax weights sum to 1) =====
#pragma unroll
        for (int nt = 0; nt < 8; ++nt) {
            const int col = d0 + nt * 16 + lo16;
            const float bcol = bl[col];
#pragma unroll
            for (int m = 0; m < 2; ++m) {
#pragma unroll
                for (int j = 0; j < 8; ++j) {
                    const int row = m * 16 + j + 8 * hiK;
                    const float xv  = XL[row * XPITCH + col];
                    const float x0v = X0[row * XPITCH + col];
                    XL[row * XPITCH + col] = xv + x0v * (bcol + acc2[m][nt][j]);
                }
            }
        }
        __syncthreads();
    }

    // ---- store result tile ----
    for (int i = tid; i < B_ROWS * (D_DIM / 4); i += THREADS) {
        const int r  = i >> 8;
        const int c4 = i & 255;
        const float4 v4 = *(const float4*)&XL[r * XPITCH + c4 * 4];
        ((float4*)(out + (size_t)(rowBase + r) * D_DIM))[c4] = v4;
    }
}

extern "C" void kernel_launch(void* const* d_in, const int* in_sizes, int n_in,
                              void* d_out, int out_size, void* d_ws, size_t ws_size,
                              hipStream_t stream) {
    (void)in_sizes; (void)n_in; (void)d_ws; (void)ws_size; (void)out_size;
    const float* x      = (const float*)d_in[0];
    const float* U      = (const float*)d_in[1];
    const float* V      = (const float*)d_in[2];
    const float* C      = (const float*)d_in[3];
    const float* bias   = (const float*)d_in[4];
    const float* gate_w = (const float*)d_in[5];
    float* out = (float*)d_out;

    const dim3 grid(B_TOT / B_ROWS);          // 512 workgroups
    const size_t smemBytes = (size_t)SMEM_FLOATS * sizeof(float);  // ~291.5 KB < 320 KB/WGP
    crossnet_mix_kernel<<<grid, THREADS, smemBytes, stream>>>(x, U, V, C, bias, gate_w, out);
}